// DeepRLSNet_40862318854348
// MI455X (gfx1250) — compile-verified
//
#include <hip/hip_runtime.h>

typedef __attribute__((ext_vector_type(2))) float v2f;
typedef __attribute__((ext_vector_type(8))) float v8f;

#define RLS_B     64
#define RLS_TAPS  64
#define RLS_N     2000

// CDNA5 async global->LDS copy (ASYNCcnt-tracked DMA, no VGPR data path).
// One lane moves 16 bytes; 16 lanes move a full 256B x-row.
__device__ __forceinline__ void async_row_to_lds_b128(const float* gsrc, unsigned lds_off) {
    asm volatile("global_load_async_to_lds_b128 %0, %1, off"
                 :: "v"(lds_off), "v"(gsrc)
                 : "memory");
}
__device__ __forceinline__ void wait_asynccnt_zero() {
    asm volatile("s_wait_asynccnt 0x0" ::: "memory");
}

// One workgroup per batch. 4 waves (wave32). Wave bi owns P rows 16*bi..16*bi+15
// as four 16x16 WMMA-C/D-layout tiles (tile (bi, t), t = 0..3).
// C/D layout (f32 16x16): lane l, vgpr v -> row i = v + 8*(l>>4), col j = l&15.
__launch_bounds__(128, 1)
__global__ void rls_scan_kernel(const float* __restrict__ x_seq,
                                const float* __restrict__ d_seq,
                                const float* __restrict__ lambdas,
                                float* __restrict__ y_out,
                                float* __restrict__ w_out)
{
    __shared__ alignas(16) float x_lds[2][RLS_TAPS];   // double buffer, async-filled
    __shared__ float Px_lds[RLS_TAPS];
    __shared__ float g_lds[RLS_TAPS];
    __shared__ float w_lds[RLS_TAPS];

    const int tid  = threadIdx.x;
    const int lane = tid & 31;
    const int wv   = tid >> 5;      // tile-row (bi) this wave owns: 0..3
    const int colj = lane & 15;     // column within a tile
    const int half = lane >> 4;     // 0: rows 0-7 of tile, 1: rows 8-15
    const int b    = blockIdx.x;    // batch index

    const float* xb = x_seq + (size_t)b * RLS_N * RLS_TAPS;
    const float* db = d_seq + (size_t)b * RLS_N;

    // ---- P0 = Identity, directly in WMMA accumulator layout ----
    v8f P[4];
#pragma unroll
    for (int t = 0; t < 4; ++t) {
        v8f p;
#pragma unroll
        for (int v = 0; v < 8; ++v) {
            const int i_local = v + 8 * half;                  // row within tile
            p[v] = (wv == t && i_local == colj) ? 1.0f : 0.0f; // diagonal tile only
        }
        P[t] = p;
    }

    if (tid < RLS_TAPS) w_lds[tid] = 0.0f;

    // ---- prologue: DMA x row 0 into buffer 0 (wave 0, lanes 0..15) ----
    if (tid < 16) {
        async_row_to_lds_b128(xb + (size_t)tid * 4,
                              (unsigned)(size_t)&x_lds[0][tid * 4]);
    }

    for (int n = 0; n < RLS_N; ++n) {
        // current buffer's DMA complete (only wave 0 has outstanding ASYNCcnt;
        // for the other waves this wait is a no-op), then publish block-wide.
        wait_asynccnt_zero();
        __syncthreads();   // B1: x[n] in LDS; previous step's P-update done

        const float* xcur = x_lds[n & 1];

        // queue next row's DMA immediately — overlaps this entire step's compute
        if (tid < 16 && (n + 1) < RLS_N) {
            async_row_to_lds_b128(xb + (size_t)(n + 1) * RLS_TAPS + tid * 4,
                                  (unsigned)(size_t)&x_lds[(n + 1) & 1][tid * 4]);
        }
        // deep prefetch to warm L2 beyond the 1-step async pipeline
        if (tid < 8 && (n + 16) < RLS_N)
            __builtin_prefetch(&xb[(size_t)(n + 16) * RLS_TAPS + tid * 8], 0, 0);

        // ---- Phase A: Px = P @ x for this wave's 16 rows (P stays in VGPRs) ----
        float acc[8];
        {
            const float xj0 = xcur[ 0 + colj];
            const float xj1 = xcur[16 + colj];
            const float xj2 = xcur[32 + colj];
            const float xj3 = xcur[48 + colj];
#pragma unroll
            for (int v = 0; v < 8; ++v)
                acc[v] = P[0][v]*xj0 + P[1][v]*xj1 + P[2][v]*xj2 + P[3][v]*xj3;
        }
        // butterfly-reduce across the 16 column lanes (bits 0..3 of lane id)
#pragma unroll
        for (int m = 1; m <= 8; m <<= 1) {
#pragma unroll
            for (int v = 0; v < 8; ++v)
                acc[v] += __shfl_xor(acc[v], m, 32);
        }
        if (colj == 0) {
#pragma unroll
            for (int v = 0; v < 8; ++v)
                Px_lds[16 * wv + 8 * half + v] = acc[v];
        }
        __syncthreads();   // B2: Px visible to all waves

        float lam = lambdas[n];                 // uniform -> s_load
        lam = fminf(fmaxf(lam, 1.0e-4f), 0.9999f);
        const float rlam = 1.0f / lam;

        // ---- Phase B (waves 0,1): denom = lam + x.Px ; y_hat = w.x ; e = d - y_hat
        float denom = 1.0f, e = 0.0f;
        if (wv < 2) {
            const float dcur = db[n];           // uniform -> s_load
            float s1 = xcur[lane] * Px_lds[lane] + xcur[lane + 32] * Px_lds[lane + 32];
            float s2 = w_lds[lane] * xcur[lane]  + w_lds[lane + 32] * xcur[lane + 32];
#pragma unroll
            for (int m = 1; m <= 16; m <<= 1) {
                s1 += __shfl_xor(s1, m, 32);
                s2 += __shfl_xor(s2, m, 32);
            }
            denom = lam + s1;
            e     = dcur - s2;
            if (tid == 0) y_out[(size_t)b * RLS_N + n] = s2;   // y_hat uses OLD w
        }
        __syncthreads();   // B3: all reads of w_lds finished

        if (tid < RLS_TAPS) {
            const float g = Px_lds[tid] / denom;
            g_lds[tid] = g;
            w_lds[tid] += g * e;
        }
        __syncthreads();   // B4: g ready for the WMMA update

        // ---- Phase C: P = (P - g * Px^T) * (1/lam), rank-1 via v_wmma_f32_16x16x4_f32
        // A (16x4) layout: vgpr0 lanes0-15 = (M=lane, K=0); lanes16-31 = (M, K=2); vgpr1 = K=1/3.
        // Only K=0 column used: A[:,0] = -g for this wave's 16 rows.
        v2f Am;
        Am[0] = (lane < 16) ? -g_lds[16 * wv + lane] : 0.0f;
        Am[1] = 0.0f;
#pragma unroll
        for (int t = 0; t < 4; ++t) {
            // B (4x16): only K=0 row used: B[0,:] = Px over this tile's 16 columns
            // (xP == Px since P remains exactly symmetric under this update).
            v2f Bm;
            Bm[0] = (lane < 16) ? Px_lds[16 * t + lane] : 0.0f;
            Bm[1] = 0.0f;
            P[t] = __builtin_amdgcn_wmma_f32_16x16x4_f32(
                       false, Am, false, Bm, (short)0, P[t], false, false);
#pragma unroll
            for (int v = 0; v < 8; ++v)
                P[t][v] *= rlam;               // dual-issue v_mul, co-executes with WMMA
        }
        // next iteration's B1 barrier orders this against the next Px_lds overwrite
    }

    if (tid < RLS_TAPS)
        w_out[(size_t)b * RLS_TAPS + tid] = w_lds[tid];
}

extern "C" void kernel_launch(void* const* d_in, const int* in_sizes, int n_in,
                              void* d_out, int out_size, void* d_ws, size_t ws_size,
                              hipStream_t stream) {
    (void)in_sizes; (void)n_in; (void)out_size; (void)d_ws; (void)ws_size;
    const float* x_seq   = (const float*)d_in[0];   // (64, 2000, 64) f32
    const float* d_seq   = (const float*)d_in[1];   // (64, 2000)     f32
    const float* lambdas = (const float*)d_in[2];   // (2000,)        f32

    float* y_out = (float*)d_out;                         // (64, 2000) flat
    float* w_out = (float*)d_out + (size_t)RLS_B * RLS_N; // (64, 64) flat

    rls_scan_kernel<<<RLS_B, 128, 0, stream>>>(x_seq, d_seq, lambdas, y_out, w_out);
}